// vertex_sampler_6837587935505
// MI455X (gfx1250) — compile-verified
//
#include <hip/hip_runtime.h>
#include <stdint.h>

// Problem constants from the reference:
// x: (16, 256, 128, 128) f32, vertexs: (16, 90, 2) int, out: (16, 256, 90) f32
#define B_    16
#define C_    256
#define H_    128
#define W_    128
#define HW_   (H_ * W_)
#define NR_   90
#define CPB   8                    // channels per block
#define CHUNKS (C_ / CPB)          // 32
#define NBLK  (B_ * CHUNKS)        // 512 blocks -> 4096 wave32s in flight
#define TPB   256                  // 8 wave32 per block
#define TOTAL (CPB * NR_)          // 720 elements per block
#define NITER ((TOTAL + TPB - 1) / TPB)  // 3 predicated iterations

#ifndef __has_builtin
#define __has_builtin(x) 0
#endif

#if defined(__HIP_DEVICE_COMPILE__) && \
    __has_builtin(__builtin_amdgcn_tensor_load_to_lds) && \
    __has_builtin(__builtin_amdgcn_s_wait_tensorcnt)
#define USE_TDM 1
#else
#define USE_TDM 0
#endif

typedef unsigned int u32;
typedef unsigned int v4u __attribute__((ext_vector_type(4)));
typedef int          v4i __attribute__((ext_vector_type(4)));
typedef int          v8i __attribute__((ext_vector_type(8)));

__global__ __launch_bounds__(TPB)
void vertex_gather_gfx1250(const float* __restrict__ x,
                           const int*   __restrict__ vtx,
                           float*       __restrict__ out) {
    // Per-batch index table staged in LDS: 90 (i,j) int pairs = 180 dwords.
    __shared__ int s_idx[NR_ * 2];

    const int b     = (int)blockIdx.x / CHUNKS;
    const int chunk = (int)blockIdx.x % CHUNKS;
    const int c0    = chunk * CPB;
    const int tid   = (int)threadIdx.x;

#if USE_TDM
    // Stage the 720-byte index table with the Tensor Data Mover (async DMA
    // global -> LDS), issued once by wave 0. 1-D descriptor:
    //   data_size = 4 B, tensor_dim0 = tile_dim0 = 180, dim1 = 1, type = 2.
    if (tid < 32) {
        const uint64_t gaddr = (uint64_t)(uintptr_t)(vtx + (size_t)b * NR_ * 2);
        const u32 lds_addr   = (u32)(uintptr_t)(void*)s_idx;  // low 32b = LDS offset
        const u32 nd = NR_ * 2;  // 180 dwords

        v4u g0;
        g0.x = 1u;                                   // count=1 (valid), user desc
        g0.y = lds_addr;                             // lds_addr [63:32]
        g0.z = (u32)gaddr;                           // global_addr lo
        g0.w = (u32)(gaddr >> 32) | (2u << 30);      // global_addr hi | type=2

        v8i g1;
        g1.s0 = (int)(2u << 16);                     // data_size=2 (4 bytes)
        g1.s1 = (int)(nd << 16);                     // tensor_dim0[15:0] @ bits 63:48
        g1.s2 = (int)(1u << 16);                     // tensor_dim1=1     @ bits 95:80
        g1.s3 = (int)(nd << 16);                     // tile_dim0=180     @ bits 127:112
        g1.s4 = 1;                                   // tile_dim1=1
        g1.s5 = (int)nd;                             // tensor_dim0_stride=180
        g1.s6 = 0;
        g1.s7 = 0;

        v4i gz = {0, 0, 0, 0};
#if __clang_major__ >= 23
        v8i gz8 = {0, 0, 0, 0, 0, 0, 0, 0};
        __builtin_amdgcn_tensor_load_to_lds(g0, g1, gz, gz, gz8, 0);
#else
        __builtin_amdgcn_tensor_load_to_lds(g0, g1, gz, gz, 0);
#endif
        __builtin_amdgcn_s_wait_tensorcnt(0);        // TENSORcnt == 0
    }
    // Compiler must not assume s_idx is unwritten (TDM wrote it).
    asm volatile("" ::: "memory");
#else
    // Fallback: plain global -> LDS staging.
    for (int t = tid; t < NR_ * 2; t += TPB)
        s_idx[t] = vtx[(size_t)b * NR_ * 2 + t];
#endif
    __syncthreads();

    const float* xb = x + ((size_t)b * C_ + c0) * HW_;
    float*       ob = out + ((size_t)b * C_ + c0) * NR_;

    // 8 channels x 90 rays = 720 elements per block; e = c_local*90 + r,
    // r fastest -> stores are contiguous runs of 90 floats (coalesced).
    // Fixed trip count (3) + full unroll -> the three independent gather
    // loads cluster before any wait, maximizing loads in flight per lane.
    const int2* __restrict__ s_pair = (const int2*)s_idx;

    float  val[NITER];
    size_t oofs[NITER];
    bool   act[NITER];

#pragma unroll
    for (int k = 0; k < NITER; ++k) {
        const int e = tid + k * TPB;
        act[k] = (e < TOTAL);
        const int ek = act[k] ? e : 0;
        const int c  = ek / NR_;
        const int r  = ek - c * NR_;
        const int2 p = s_pair[r];
        const int ij = p.x * W_ + p.y;
        val[k]  = xb[(size_t)c * HW_ + (size_t)ij];
        oofs[k] = (size_t)c * NR_ + (size_t)r;
    }

#pragma unroll
    for (int k = 0; k < NITER; ++k) {
        if (act[k])
            __builtin_nontemporal_store(val[k], &ob[oofs[k]]);
    }
}

extern "C" void kernel_launch(void* const* d_in, const int* in_sizes, int n_in,
                              void* d_out, int out_size, void* d_ws, size_t ws_size,
                              hipStream_t stream) {
    (void)in_sizes; (void)n_in; (void)out_size; (void)d_ws; (void)ws_size;
    const float* x   = (const float*)d_in[0];
    const int*   vtx = (const int*)d_in[1];
    float*       out = (float*)d_out;

    dim3 grid(NBLK), block(TPB);
    hipLaunchKernelGGL(vertex_gather_gfx1250, grid, block, 0, stream, x, vtx, out);
}